// VolatilityLoss_67534065762413
// MI455X (gfx1250) — compile-verified
//
#include <hip/hip_runtime.h>
#include <math.h>

// VolatilityLoss on MI455X (gfx1250, wave32).
// predictions, targets: [512, 8192] fp32 -> scalar fp32 mean of
// (slidingStd3(pred) - slidingStd3(targ))^2 over 512 x 8190 windows.
//
// Bandwidth-bound: 33.5 MB unique reads, ~1.4 us floor at 23.3 TB/s.
// Structure: per-block LDS staging via GLOBAL_LOAD_ASYNC_TO_LDS_B128
// (ASYNCcnt), window math from LDS (raw v_sqrt_f32 - operands are
// guaranteed normal and >= 0), exact-fp32 WMMA 16x16x4 accumulate
// (A = ones), deterministic two-pass reduction.

typedef __attribute__((ext_vector_type(2))) float v2f;
typedef __attribute__((ext_vector_type(8))) float v8f;
typedef __attribute__((ext_vector_type(4))) int   v4i;

#define B_ROWS   512
#define S_LEN    8192
#define L_LEN    (S_LEN - 2)            // 8190 windows per row
#define N_WIN    (B_ROWS * L_LEN)       // 4,193,280 windows
#define CHUNK    2048                   // windows per block
#define CPR      4                      // chunks per row (4*2048 >= 8190)
#define NBLOCKS  (B_ROWS * CPR)         // 2048
#define NTHREADS 256
#define ELEMS    (CHUNK + 4)            // LDS floats per array (16B multiple)

#if defined(__HIP_DEVICE_COMPILE__) && __has_builtin(__builtin_amdgcn_global_load_async_to_lds_b128)
#define USE_ASYNC_BUILTIN 1
#else
#define USE_ASYNC_BUILTIN 0
#endif
#if defined(__HIP_DEVICE_COMPILE__) && __has_builtin(__builtin_amdgcn_s_wait_asynccnt)
#define USE_WAIT_BUILTIN 1
#else
#define USE_WAIT_BUILTIN 0
#endif

// Copy 16 bytes global -> LDS asynchronously (per-lane addresses).
__device__ __forceinline__ void async_copy16(const float* gsrc, float* ldst) {
#if defined(__HIP_DEVICE_COMPILE__)
#if USE_ASYNC_BUILTIN
    typedef __attribute__((address_space(1))) v4i* gptr_t;
    typedef __attribute__((address_space(3))) v4i* lptr_t;
    __builtin_amdgcn_global_load_async_to_lds_b128(
        (gptr_t)gsrc, (lptr_t)ldst, /*offset=*/0, /*cpol=*/0);
#else
    unsigned lds_off =
        (unsigned)(uintptr_t)(__attribute__((address_space(3))) void*)ldst;
    unsigned long long ga = (unsigned long long)(uintptr_t)gsrc;
    asm volatile("global_load_async_to_lds_b128 %0, %1, off"
                 :: "v"(lds_off), "v"(ga) : "memory");
#endif
#else
    (void)gsrc; (void)ldst;
#endif
}

__device__ __forceinline__ void wait_async0() {
#if defined(__HIP_DEVICE_COMPILE__)
#if USE_WAIT_BUILTIN
    __builtin_amdgcn_s_wait_asynccnt(0);
#else
    asm volatile("s_wait_asynccnt 0x0" ::: "memory");
#endif
#endif
}

// Fast sqrt: raw v_sqrt_f32 (~1 ulp). Operands here are 0.5*sum((x-m)^2) for
// ~N(0,1) data: always >= 0 and comfortably normal, so the IEEE denormal
// fix-up sequence libm's sqrtf drags in is dead weight.
__device__ __forceinline__ float fast_sqrtf(float x) {
#if defined(__HIP_DEVICE_COMPILE__)
    return __builtin_amdgcn_sqrtf(x);
#else
    return sqrtf(x);
#endif
}

// std(window of 3, ddof=1) difference squared, elements from LDS.
__device__ __forceinline__ float win_from_lds(const float* __restrict__ lp,
                                              const float* __restrict__ lt,
                                              int j) {
    float p0 = lp[j], p1 = lp[j + 1], p2 = lp[j + 2];
    float mp = (p0 + p1 + p2) * (1.0f / 3.0f);
    float a0 = p0 - mp, a1 = p1 - mp, a2 = p2 - mp;
    float sp = fast_sqrtf((a0 * a0 + a1 * a1 + a2 * a2) * 0.5f);

    float t0 = lt[j], t1 = lt[j + 1], t2 = lt[j + 2];
    float mt = (t0 + t1 + t2) * (1.0f / 3.0f);
    float b0 = t0 - mt, b1 = t1 - mt, b2 = t2 - mt;
    float st = fast_sqrtf((b0 * b0 + b1 * b1 + b2 * b2) * 0.5f);

    float d = sp - st;
    return d * d;
}

__global__ void __launch_bounds__(NTHREADS)
vol_loss_main(const float* __restrict__ P, const float* __restrict__ T,
              float* __restrict__ partials) {
    __shared__ __align__(16) float lp[ELEMS];
    __shared__ __align__(16) float lt[ELEMS];

    const int row = blockIdx.y;
    const int c0  = blockIdx.x * CHUNK;                  // element col offset
    const long long gbase = (long long)row * S_LEN + c0; // affine, no div/mod

    // Elements needed from this row; clip at row end (no OOB global reads).
    const int elems = min(CHUNK + 2, S_LEN - c0);        // 2050 or 2048
    const int n16   = (elems + 3) >> 2;                  // 16B segments

    for (int seg = threadIdx.x; seg < n16; seg += NTHREADS) {
        async_copy16(P + gbase + seg * 4, lp + seg * 4);
        async_copy16(T + gbase + seg * 4, lt + seg * 4);
    }
    wait_async0();        // this wave's ASYNCcnt -> 0
    __syncthreads();      // all waves' LDS writes visible

    const int nwin = min(CHUNK, L_LEN - c0);             // 2048 or 2046

    // Exact-fp32 accumulate: D = ones(16x4) x B(4x16) + C, all D rows equal.
    v8f acc = {};
    v2f aOnes;
    aOnes[0] = 1.0f;
    aOnes[1] = 1.0f;

    // Uniform trip count; tail handled by cndmask select (EXEC stays all-1s,
    // as WMMA requires). LDS reads are always in-bounds (max j+2 = 2049).
    #pragma unroll
    for (int k = 0; k < 8; k += 2) {
        const int j0 = threadIdx.x + k * NTHREADS;
        const int j1 = j0 + NTHREADS;
        float v0 = win_from_lds(lp, lt, j0);
        float v1 = win_from_lds(lp, lt, j1);
        v0 = (j0 < nwin) ? v0 : 0.0f;
        v1 = (j1 < nwin) ? v1 : 0.0f;
        v2f b;
        b[0] = v0;
        b[1] = v1;
        acc = __builtin_amdgcn_wmma_f32_16x16x4_f32(
            /*neg_a=*/false, aOnes, /*neg_b=*/false, b,
            /*c_mod=*/(short)0, acc, /*reuse_a=*/false, /*reuse_b=*/false);
    }

    // acc[0]: lanes 0..15 = row M=0, lanes 16..31 = row M=8 (identical rows,
    // A all ones). Sum over 32 lanes double-counts -> * 0.5.
    float s = acc[0];
    #pragma unroll
    for (int off = 16; off > 0; off >>= 1)
        s += __shfl_xor(s, off, 32);
    s *= 0.5f;

    __shared__ float red[NTHREADS / 32];
    const int lane = threadIdx.x & 31;
    const int wave = threadIdx.x >> 5;
    if (lane == 0) red[wave] = s;
    __syncthreads();
    if (threadIdx.x == 0) {
        float tot = 0.0f;
        #pragma unroll
        for (int i = 0; i < NTHREADS / 32; ++i) tot += red[i];
        partials[blockIdx.y * CPR + blockIdx.x] = tot;  // rewritten every call
    }
}

__global__ void __launch_bounds__(NTHREADS)
vol_loss_final(const float* __restrict__ partials, float* __restrict__ out) {
    // Single block; fixed summation order -> deterministic.
    float s = 0.0f;
    for (int i = threadIdx.x; i < NBLOCKS; i += NTHREADS)
        s += partials[i];
    #pragma unroll
    for (int off = 16; off > 0; off >>= 1)
        s += __shfl_xor(s, off, 32);

    __shared__ float red[NTHREADS / 32];
    const int lane = threadIdx.x & 31;
    const int wave = threadIdx.x >> 5;
    if (lane == 0) red[wave] = s;
    __syncthreads();
    if (threadIdx.x == 0) {
        float tot = 0.0f;
        #pragma unroll
        for (int i = 0; i < NTHREADS / 32; ++i) tot += red[i];
        out[0] = tot / (float)N_WIN;
    }
}

extern "C" void kernel_launch(void* const* d_in, const int* in_sizes, int n_in,
                              void* d_out, int out_size, void* d_ws, size_t ws_size,
                              hipStream_t stream) {
    const float* P = (const float*)d_in[0];   // predictions [512, 8192] f32
    const float* T = (const float*)d_in[1];   // targets     [512, 8192] f32
    float* out      = (float*)d_out;          // scalar f32
    float* partials = (float*)d_ws;           // NBLOCKS floats of scratch

    dim3 grid(CPR, B_ROWS);
    vol_loss_main<<<grid, NTHREADS, 0, stream>>>(P, T, partials);
    vol_loss_final<<<1, NTHREADS, 0, stream>>>(partials, out);
}